// TSC_input_18751827214348
// MI455X (gfx1250) — compile-verified
//
#include <hip/hip_runtime.h>

typedef __attribute__((ext_vector_type(2))) float v2f;
typedef __attribute__((ext_vector_type(4))) float v4f;
typedef __attribute__((ext_vector_type(8))) float v8f;

#define BATCH    64
#define NBLK     1024              // 1024-element blocks per data row
#define MROWS    (BATCH * NBLK)    // 65536 GEMM rows
#define KDIM     1024
#define NDIM     512               // 511 real outputs + 1 zero pad column
#define OUT_LEN  524287            // 2*(L>>2)-1
#define MTILE    64                // rows per workgroup (4x less W traffic)
#define KC       64                // K chunk staged in LDS
#define LDS_PITCH 68               // 64 + 4 pad -> conflict-free fragment reads

// ---------------------------------------------------------------------------
// Dense tap value for (K,n): column n (0..510): e=log2(n+1), scale s=10-e,
// w_width=2^s, k_local=2^e, t=n-(k_local-1).
// W[K][n] = (K/w_width==t) ? weights[w_width-1 + K%w_width] : 0.
// ---------------------------------------------------------------------------
__device__ __forceinline__ float tap_value(const float* __restrict__ weights,
                                           int K, int n) {
    if (n >= 511) return 0.0f;              // pad column
    int e       = 31 - __clz(n + 1);        // 0..8
    int k_local = 1 << e;
    int w_width = KDIM >> e;                // 2^s
    int t       = n - (k_local - 1);
    if ((K >> (10 - e)) != t) return 0.0f;
    return weights[w_width - 1 + (K & (w_width - 1))];
}

// ---------------------------------------------------------------------------
// Build W pre-swizzled into WMMA B-fragment order:
// float index = ((o*2 + h)*512 + n)*4 + c, with o=K-octet (0..127), h=lane
// half, c selects (quad0 K=8o+2h, 8o+2h+1, quad1 K=8o+4+2h, 8o+4+2h+1).
// A single b128 per lane then feeds two v_wmma_f32_16x16x4_f32.
// ---------------------------------------------------------------------------
__global__ void TSC_build_wpack(const float* __restrict__ weights,
                                float* __restrict__ Wp) {
    int idx = blockIdx.x * blockDim.x + threadIdx.x;
    if (idx >= KDIM * NDIM) return;
    int c = idx & 3;
    int n = (idx >> 2) & 511;
    int h = (idx >> 11) & 1;
    int o = idx >> 12;                       // 0..127
    int K = 8 * o + 2 * h + ((c & 2) << 1) + (c & 1);
    Wp[idx] = tap_value(weights, K, n);
}

// ---------------------------------------------------------------------------
// Zero the first 1023 columns of each output row (never written by any scale).
// ---------------------------------------------------------------------------
__global__ void TSC_zero_prefix(float* __restrict__ out) {
    int idx = blockIdx.x * blockDim.x + threadIdx.x;
    if (idx >= BATCH * 1023) return;
    int b = idx / 1023;
    int c = idx - b * 1023;
    out[(size_t)b * OUT_LEN + c] = 0.0f;
}

// ---------------------------------------------------------------------------
// C[65536x512] = A[65536x1024] * W[1024x512] via v_wmma_f32_16x16x4_f32.
// Workgroup: 64-row strip, 8 waves = 4 M-groups x 2 N-groups; each wave owns
// a 16x256 C strip (16 accumulator tiles).  Fused bias + scatter epilogue.
// ---------------------------------------------------------------------------
__global__ __launch_bounds__(256)
void TSC_msconv_wmma(const float* __restrict__ A,
                     const float* __restrict__ Wp,
                     const float* __restrict__ bias,
                     float* __restrict__ out) {
    __shared__ float lds_a[MTILE * LDS_PITCH];

    const int tid  = threadIdx.x;
    const int lane = tid & 31;
    const int wv   = tid >> 5;          // wave 0..7
    const int half = lane >> 4;         // lane-half selects K 0/1 vs 2/3
    const int l15  = lane & 15;
    const int mg   = wv & 3;            // M group: rows [16*mg, 16*mg+16)
    const int ng   = wv >> 2;           // N group: cols [256*ng, 256*ng+256)
    const int r0   = blockIdx.x * MTILE;

    v8f acc[16];
#pragma unroll
    for (int t = 0; t < 16; ++t) acc[t] = (v8f){};

    // cooperative-load: 256 threads x 16 floats = 64x64 chunk
    const int mld = tid >> 2;           // row 0..63
    const int kld = (tid & 3) << 4;     // k offset 0,16,32,48

    const int arow  = (mg * 16 + l15) * LDS_PITCH;
    const int nbase = (ng << 8) + l15;  // wave's first column + lane offset

    for (int kc = 0; kc < KDIM; kc += KC) {
        {
            const float* g = A + (size_t)(r0 + mld) * KDIM + kc + kld;
            float4 p0 = *(const float4*)(g);
            float4 p1 = *(const float4*)(g + 4);
            float4 p2 = *(const float4*)(g + 8);
            float4 p3 = *(const float4*)(g + 12);
            float* d = &lds_a[mld * LDS_PITCH + kld];
            *(float4*)(d)      = p0;
            *(float4*)(d + 4)  = p1;
            *(float4*)(d + 8)  = p2;
            *(float4*)(d + 12) = p3;
        }
        __syncthreads();

        for (int kk = 0; kk < KC; kk += 8) {
            // A fragments (quad 0 and quad 1 of this K-octet) from LDS
            const int ka = kk + (half << 1);
            v2f a0, a1;
            a0.x = lds_a[arow + ka];
            a0.y = lds_a[arow + ka + 1];
            a1.x = lds_a[arow + ka + 4];
            a1.y = lds_a[arow + ka + 5];

            // one b128 per tile feeds two WMMAs
            const int o = (kc + kk) >> 3;
            const v4f* wp = (const v4f*)Wp + ((o * 2 + half) * NDIM + nbase);
#pragma unroll
            for (int t = 0; t < 16; ++t) {
                v4f bf = wp[t * 16];
                v2f b0 = bf.xy;          // quad 0 fragment
                v2f b1 = bf.zw;          // quad 1 fragment
                acc[t] = __builtin_amdgcn_wmma_f32_16x16x4_f32(
                    false, a0, false, b0, (short)0, acc[t], false, false);
                acc[t] = __builtin_amdgcn_wmma_f32_16x16x4_f32(
                    false, a1, false, b1, (short)0, acc[t], false, false);
            }
        }
        __syncthreads();
    }

    // Epilogue: bias + scatter.  C layout: VGPR j = row j (lanes 0-15) or
    // row j+8 (lanes 16-31); column = lane&15 within tile.
    const int rbase   = r0 + mg * 16;       // first global row of wave's strip
    const int brow    = rbase >> 10;        // batch index (strip never crosses)
    const int blkbase = rbase & (NBLK - 1); // block index of strip row 0

#pragma unroll
    for (int t = 0; t < 16; ++t) {
        const int n = (ng << 8) + (t << 4) + l15;    // 0..511
        if (n == 511) continue;                      // pad column
        const int   e       = 31 - __clz(n + 1);     // 0..8
        const int   k_local = 1 << e;
        const float bv      = bias[9 - e];           // bias[s-1], s = 10-e
        const v8f   av      = acc[t];
#pragma unroll
        for (int j = 0; j < 8; ++j) {
            const int m   = j + (half << 3);
            const int blk = blkbase + m;
            const size_t col = (size_t)k_local * (blk + 1023) + n;
            out[(size_t)brow * OUT_LEN + col] = av[j] + bv;
        }
    }
}

extern "C" void kernel_launch(void* const* d_in, const int* in_sizes, int n_in,
                              void* d_out, int out_size, void* d_ws, size_t ws_size,
                              hipStream_t stream) {
    const float* data    = (const float*)d_in[0];   // (64, 2^20)
    const float* weights = (const float*)d_in[1];   // (2047,)
    const float* bias    = (const float*)d_in[2];   // (10,)
    float*       out     = (float*)d_out;           // (64, 524287)
    float*       Wp      = (float*)d_ws;            // 1024*512 floats = 2 MB

    TSC_build_wpack<<<(KDIM * NDIM + 255) / 256, 256, 0, stream>>>(weights, Wp);
    TSC_zero_prefix<<<(BATCH * 1023 + 255) / 256, 256, 0, stream>>>(out);
    TSC_msconv_wmma<<<MROWS / MTILE, 256, 0, stream>>>(data, Wp, bias, out);
}